// TemporalCausalAttention_4209067950434
// MI455X (gfx1250) — compile-verified
//
#include <hip/hip_runtime.h>
#include <hip/hip_bf16.h>

// ---------------------------------------------------------------------------
// TemporalCausalAttention for MI455X (gfx1250, wave32, WMMA).
// All heavy contractions run on V_WMMA_F32_16X16X4_F32 (exact fp32 matrix op).
// Problem is HBM-bound (~1GB traffic @ 23.3TB/s >> 29GFLOP on matrix pipes),
// so the design minimizes materialized intermediates:
//   rproj (134MB) eliminated via  q·rproj = R · (q·Wr)
//   Cij   (134MB) eliminated via  Σj w·(R·W3ᵀ) = (Σj w·R)·W3ᵀ
// ---------------------------------------------------------------------------

typedef float v2f __attribute__((ext_vector_type(2)));
typedef float v8f __attribute__((ext_vector_type(8)));

// Only meaningful on the device pass; host pass has no amdgcn builtins.
#if defined(__HIP_DEVICE_COMPILE__) && !__has_builtin(__builtin_amdgcn_wmma_f32_16x16x4_f32)
#error "missing __builtin_amdgcn_wmma_f32_16x16x4_f32 for gfx1250"
#endif

#define BMT 64
#define BNT 64
#define BKT 32

// Generic batched GEMM:  C[m,n] (+)= sum_k A[m,k] * W(n,k)  (+ bias[n])
//   A element  : Ab + m*aSm + k            (K must be contiguous in A)
//   W element  : Wb + n*wSn + k*wSk        (either wSk==1 [NT] or wSn==1 [NN])
//   C element  : Cb + m*cSm + n*cSn
//   batch z    : z0=z/zdiv, z1=z%zdiv;  base += z0*S0 + z1*S1 per operand
// Requirements: M % 64 == 0, K % 32 == 0, N % 4 == 0 (N<64 handled by guards).
__global__ __launch_bounds__(256) void gemm_f32_wmma(
    const float* __restrict__ A, long long aS0, long long aS1, int aSm,
    const float* __restrict__ W, long long wS0, long long wS1, int wSn, int wSk,
    const float* __restrict__ bias,
    float* __restrict__ C, long long cS0, long long cS1, int cSm, int cSn,
    int M, int N, int K, int zdiv, int accumulate)
{
    __shared__ float As[BMT][BKT + 1];
    __shared__ float Ws[BKT][BNT + 1];

    const int tid = threadIdx.x;
    const int z   = blockIdx.z;
    const int z0  = z / zdiv, z1 = z % zdiv;
    const float* Ab = A + z0 * aS0 + z1 * aS1;
    const float* Wb = W + z0 * wS0 + z1 * wS1;
    float*       Cb = C + z0 * cS0 + z1 * cS1;

    const int m0 = blockIdx.x * BMT;
    const int n0 = blockIdx.y * BNT;

    const int wid  = tid >> 5;
    const int lane = tid & 31;
    const int half = lane >> 4;       // wave32: lanes 0-15 hold K{0,1}, 16-31 hold K{2,3}
    const int l16  = lane & 15;
    const int wm   = (wid & 3) * 16;  // 4 waves stacked in M
    const int wn   = (wid >> 2) * 32; // 2 waves stacked in N, 2 subtiles each

    v8f acc0 = {0.f,0.f,0.f,0.f,0.f,0.f,0.f,0.f};
    v8f acc1 = acc0;

    for (int k0 = 0; k0 < K; k0 += BKT) {
        // ---- stage A tile (64x32), coalesced float4 along K ----
        #pragma unroll
        for (int p = 0; p < 2; ++p) {
            int f4 = tid + p * 256;
            int kc = (f4 & 7) * 4;
            int r  = f4 >> 3;
            const float4 v = *(const float4*)(Ab + (long long)(m0 + r) * aSm + (k0 + kc));
            As[r][kc+0] = v.x; As[r][kc+1] = v.y; As[r][kc+2] = v.z; As[r][kc+3] = v.w;
        }
        // ---- stage W tile (store as Ws[k][n]) ----
        if (wSk == 1) {            // NT: K contiguous in W
            #pragma unroll
            for (int p = 0; p < 2; ++p) {
                int f4 = tid + p * 256;
                int kc = (f4 & 7) * 4;
                int n  = f4 >> 3;
                float4 v = make_float4(0.f, 0.f, 0.f, 0.f);
                if (n0 + n < N)
                    v = *(const float4*)(Wb + (long long)(n0 + n) * wSn + (k0 + kc));
                Ws[kc+0][n] = v.x; Ws[kc+1][n] = v.y; Ws[kc+2][n] = v.z; Ws[kc+3][n] = v.w;
            }
        } else {                   // NN: N contiguous in W
            #pragma unroll
            for (int p = 0; p < 2; ++p) {
                int f4 = tid + p * 256;
                int n4 = (f4 & 15) * 4;
                int kk = f4 >> 4;
                float4 v = make_float4(0.f, 0.f, 0.f, 0.f);
                if (n0 + n4 < N)
                    v = *(const float4*)(Wb + (long long)(k0 + kk) * wSk + (n0 + n4));
                Ws[kk][n4+0] = v.x; Ws[kk][n4+1] = v.y; Ws[kk][n4+2] = v.z; Ws[kk][n4+3] = v.w;
            }
        }
        __syncthreads();
        // ---- 8 x (two 16x16x4 WMMAs), fragments per ISA f32 layout ----
        #pragma unroll
        for (int kk = 0; kk < BKT; kk += 4) {
            const int kr = kk + 2 * half;
            v2f a;  a.x  = As[wm + l16][kr];       a.y  = As[wm + l16][kr + 1];
            v2f b0; b0.x = Ws[kr][wn + l16];       b0.y = Ws[kr + 1][wn + l16];
            v2f b1; b1.x = Ws[kr][wn + 16 + l16];  b1.y = Ws[kr + 1][wn + 16 + l16];
            acc0 = __builtin_amdgcn_wmma_f32_16x16x4_f32(false, a, false, b0, (short)0, acc0, false, false);
            acc1 = __builtin_amdgcn_wmma_f32_16x16x4_f32(false, a, false, b1, (short)0, acc1, false, false);
        }
        __syncthreads();
    }

    // ---- epilogue: C/D layout lanes0-15 -> M=v, lanes16-31 -> M=v+8 ----
    #pragma unroll
    for (int v = 0; v < 8; ++v) {
        const int m  = m0 + wm + v + 8 * half;   // m < M guaranteed (M % 64 == 0)
        const int na = n0 + wn + l16;
        const int nb = na + 16;
        if (na < N) {
            long long idx = (long long)m * cSm + (long long)na * cSn;
            float val = acc0[v];
            if (bias)       val += bias[na];
            if (accumulate) val += Cb[idx];
            Cb[idx] = val;
        }
        if (nb < N) {
            long long idx = (long long)m * cSm + (long long)nb * cSn;
            float val = acc1[v];
            if (bias)       val += bias[nb];
            if (accumulate) val += Cb[idx];
            Cb[idx] = val;
        }
    }
}

// dsum[b,n,e] = sum_t delta_H[b,n,t,e]
__global__ __launch_bounds__(128) void sum_t_kernel(const float* __restrict__ dH,
                                                    float* __restrict__ dsum)
{
    const int bn = blockIdx.x, e = threadIdx.x;
    const float* p = dH + ((long long)bn * 64) * 128 + e;
    float s = 0.f;
    #pragma unroll 8
    for (int t = 0; t < 64; ++t) s += p[t * 128];
    dsum[(long long)bn * 128 + e] = s;
}

// out[row] = dot(X[row, 0:128], v)   (one wave32 per row, coalesced)
__global__ __launch_bounds__(256) void rowdot128_kernel(const float* __restrict__ X,
                                                        const float* __restrict__ v,
                                                        float* __restrict__ out, int rows)
{
    const int wid = threadIdx.x >> 5, lane = threadIdx.x & 31;
    const int row = blockIdx.x * 8 + wid;
    if (row >= rows) return;
    const float* x = X + (long long)row * 128;
    float s = 0.f;
    #pragma unroll
    for (int c = 0; c < 4; ++c) s += x[lane + 32 * c] * v[lane + 32 * c];
    #pragma unroll
    for (int off = 16; off; off >>= 1) s += __shfl_down(s, off, 32);
    if (lane == 0) out[row] = s;
}

// Per (b,i): scores[j,t] = (sc1[b,t,i,j] + sum_r R[b,i,j,r]*qr[b,i,t,r] + qb[b,i,t]) * scale
// softmax over t, zero row j==i, write w in (b,i,t,j) layout (K-contiguous for GEMMs).
__global__ __launch_bounds__(256) void softmax_kernel(
    const float* __restrict__ sc1, const float* __restrict__ R,
    const float* __restrict__ qr,  const float* __restrict__ qb,
    float* __restrict__ w2)
{
    __shared__ float qr_s[64][33];
    __shared__ float qb_s[64];
    const int tid = threadIdx.x;         // j
    const int bi  = blockIdx.x;          // b*256 + i
    const int b   = bi >> 8;
    const int i   = bi & 255;

    #pragma unroll
    for (int p = 0; p < 2; ++p) {        // stage qr (64x32)
        int f4 = tid + p * 256;
        int t  = f4 >> 3;
        int r4 = (f4 & 7) * 4;
        float4 v = *(const float4*)(qr + ((long long)bi * 64 + t) * 32 + r4);
        qr_s[t][r4+0] = v.x; qr_s[t][r4+1] = v.y; qr_s[t][r4+2] = v.z; qr_s[t][r4+3] = v.w;
    }
    if (tid < 64) qb_s[tid] = qb[(long long)bi * 64 + tid];

    float rr[32];
    {
        const float* rp = R + ((long long)bi * 256 + tid) * 32;
        #pragma unroll
        for (int c = 0; c < 8; ++c) {
            float4 v = *(const float4*)(rp + 4 * c);
            rr[4*c+0] = v.x; rr[4*c+1] = v.y; rr[4*c+2] = v.z; rr[4*c+3] = v.w;
        }
    }
    __syncthreads();

    const float scale = 0.08838834764831845f;   // 1/sqrt(128)
    float s[64];
    float smax = -3.0e38f;
    #pragma unroll 4
    for (int t = 0; t < 64; ++t) {
        float acc = sc1[(((long long)(b * 64 + t)) * 256 + i) * 256 + tid] + qb_s[t];
        #pragma unroll
        for (int r = 0; r < 32; ++r) acc += rr[r] * qr_s[t][r];
        acc *= scale;
        s[t] = acc;
        smax = fmaxf(smax, acc);
    }
    float sum = 0.f;
    #pragma unroll 4
    for (int t = 0; t < 64; ++t) { float e = expf(s[t] - smax); s[t] = e; sum += e; }
    float inv = 1.f / sum;
    if (tid == i) inv = 0.f;             // mask diagonal j == i
    #pragma unroll 4
    for (int t = 0; t < 64; ++t)
        w2[((long long)bi * 64 + t) * 256 + tid] = s[t] * inv;
}

// wsum[row] = sum_j w2[row, j]   (deterministic tree reduction)
__global__ __launch_bounds__(256) void wsum_kernel(const float* __restrict__ w2,
                                                   float* __restrict__ wsum)
{
    __shared__ float red[256];
    const long long row = blockIdx.x;
    red[threadIdx.x] = w2[row * 256 + threadIdx.x];
    __syncthreads();
    for (int s = 128; s > 0; s >>= 1) {
        if (threadIdx.x < s) red[threadIdx.x] += red[threadIdx.x + s];
        __syncthreads();
    }
    if (threadIdx.x == 0) wsum[row] = red[0];
}

// x = H + z_acc + Aout*wsum ; LayerNorm over D=128 ; in-place on out buffer
__global__ __launch_bounds__(128) void ln_kernel(
    const float* __restrict__ H, const float* __restrict__ Aout,
    const float* __restrict__ wsum, const float* __restrict__ gamma,
    const float* __restrict__ beta, float* __restrict__ out)
{
    __shared__ float r1[128];
    __shared__ float r2[128];
    const long long row = blockIdx.x;
    const int e = threadIdx.x;
    const long long idx = row * 128 + e;
    const float x = H[idx] + out[idx] + Aout[idx] * wsum[row];
    r1[e] = x; r2[e] = x * x;
    __syncthreads();
    for (int s = 64; s > 0; s >>= 1) {
        if (e < s) { r1[e] += r1[e + s]; r2[e] += r2[e + s]; }
        __syncthreads();
    }
    const float mu  = r1[0] * (1.f / 128.f);
    const float var = r2[0] * (1.f / 128.f) - mu * mu;
    out[idx] = (x - mu) * rsqrtf(var + 1e-5f) * gamma[e] + beta[e];
}

extern "C" void kernel_launch(void* const* d_in, const int* in_sizes, int n_in,
                              void* d_out, int out_size, void* d_ws, size_t ws_size,
                              hipStream_t stream)
{
    constexpr int B = 4, N = 256, T = 64, D = 128, DR = 32, DH = 128;
    constexpr int LDF = 3 * D + DR;   // Wf row stride = 416

    const float* H   = (const float*)d_in[0];
    const float* R   = (const float*)d_in[1];
    const float* dHp = (const float*)d_in[2];
    const float* Wq  = (const float*)d_in[3];
    const float* bq  = (const float*)d_in[4];
    const float* Wk  = (const float*)d_in[5];
    const float* bk  = (const float*)d_in[6];
    const float* Wr  = (const float*)d_in[7];
    const float* br  = (const float*)d_in[8];
    const float* Wf  = (const float*)d_in[9];
    const float* bf  = (const float*)d_in[10];
    const float* gam = (const float*)d_in[11];
    const float* bet = (const float*)d_in[12];
    float* out = (float*)d_out;

    // workspace carve-out (~287 MB total, all 16B aligned)
    float* ws = (float*)d_ws;
    size_t off = 0;
    auto alloc = [&](size_t n) { float* p = ws + off; off += n; return p; };
    float* q    = alloc((size_t)B * N * T * DH);  // 8.4M
    float* kt   = alloc((size_t)B * N * T * DH);  // 8.4M
    float* Aout = alloc((size_t)B * N * T * D);   // 8.4M  (H·W1ᵀ + bf)
    float* Bm   = alloc((size_t)B * N * T * D);   // 8.4M  (H·W2ᵀ)
    float* qr   = alloc((size_t)B * N * T * DR);  // 2.1M  (q·Wr)
    float* qb   = alloc((size_t)B * N * T);       // 65K   (q·br)
    float* dsum = alloc((size_t)B * N * D);       // 131K
    float* Ej   = alloc((size_t)B * N * D);       // 131K
    float* sc1  = alloc((size_t)B * T * N * N);   // 16.8M (qk scores, (b,t,i,j))
    float* w2   = alloc((size_t)B * N * T * N);   // 16.8M (softmax w, (b,i,t,j))
    float* WR   = alloc((size_t)B * N * T * DR);  // 2.1M  (Σj w·R)
    float* wsm  = alloc((size_t)B * N * T);       // 65K
    (void)ws_size; (void)in_sizes; (void)n_in; (void)out_size;

    const dim3 blk(256);
    const long long LNTD  = (long long)N * T * D;    // 2097152
    const long long LNTDH = (long long)N * T * DH;   // 2097152
    const long long LNTN  = (long long)N * T * N;    // 4194304
    const long long LTNN  = (long long)T * N * N;    // 4194304

    // 1) dsum = sum_t delta_H
    sum_t_kernel<<<B * N, 128, 0, stream>>>(dHp, dsum);

    // 2-5) projections from H:  q, k, Aout(+bf), Bm
    gemm_f32_wmma<<<dim3(B*N*T/64, DH/64, 1), blk, 0, stream>>>(
        H, 0, 0, D,  Wq, 0, 0, D, 1,  bq,  q, 0, 0, DH, 1,  B*N*T, DH, D, 1, 0);
    gemm_f32_wmma<<<dim3(B*N*T/64, DH/64, 1), blk, 0, stream>>>(
        H, 0, 0, D,  Wk, 0, 0, D, 1,  bk,  kt, 0, 0, DH, 1,  B*N*T, DH, D, 1, 0);
    gemm_f32_wmma<<<dim3(B*N*T/64, D/64, 1), blk, 0, stream>>>(
        H, 0, 0, D,  Wf, 0, 0, LDF, 1,  bf,  Aout, 0, 0, D, 1,  B*N*T, D, D, 1, 0);
    gemm_f32_wmma<<<dim3(B*N*T/64, D/64, 1), blk, 0, stream>>>(
        H, 0, 0, D,  Wf + D, 0, 0, LDF, 1,  nullptr,  Bm, 0, 0, D, 1,  B*N*T, D, D, 1, 0);

    // 6) Ej = dsum·W4ᵀ
    gemm_f32_wmma<<<dim3(B*N/64, D/64, 1), blk, 0, stream>>>(
        dsum, 0, 0, D,  Wf + 2*D + DR, 0, 0, LDF, 1,  nullptr,
        Ej, 0, 0, D, 1,  B*N, D, D, 1, 0);

    // 7) qr[m,r] = sum_h q[m,h]*Wr[h,r]   (W is NN: element at h*DR + r)
    gemm_f32_wmma<<<dim3(B*N*T/64, 1, 1), blk, 0, stream>>>(
        q, 0, 0, DH,  Wr, 0, 0, 1, DR,  nullptr,  qr, 0, 0, DR, 1,  B*N*T, DR, DH, 1, 0);

    // 8) qb = q·br
    rowdot128_kernel<<<B * N * T / 8, blk, 0, stream>>>(q, br, qb, B * N * T);

    // 9) term1: per (b,t): sc1[i,j] = sum_h q[b,i,t,h]*k[b,j,t,h]
    gemm_f32_wmma<<<dim3(N/64, N/64, B*T), blk, 0, stream>>>(
        q,  LNTDH, DH, T*DH,   kt, LNTDH, DH, T*DH, 1,   nullptr,
        sc1, LTNN, (long long)N*N, N, 1,   N, N, DH, T, 0);

    // 10) softmax (+ R·qr rank-32 term + qb bias term, mask, transpose to (b,i,t,j))
    softmax_kernel<<<B * N, blk, 0, stream>>>(sc1, R, qr, qb, w2);

    // 11) WR: per (b,i): WR[t,r] = sum_j w2[t,j]*R[j,r]   (R is NN: j*DR + r)
    gemm_f32_wmma<<<dim3(T/64, 1, B*N), blk, 0, stream>>>(
        w2, (long long)T*N, 0, N,   R, (long long)N*DR, 0, 1, DR,   nullptr,
        WR, (long long)T*DR, 0, DR, 1,   T, DR, N, 1, 0);

    // 12) z  = WR·W3ᵀ                  (overwrite d_out)
    gemm_f32_wmma<<<dim3(B*N*T/64, D/64, 1), blk, 0, stream>>>(
        WR, 0, 0, DR,  Wf + 2*D, 0, 0, LDF, 1,  nullptr,
        out, 0, 0, D, 1,  B*N*T, D, DR, 1, 0);

    // 13) z += per (b,i): sum_j w2[t,j]*Ej[b,j,e]   (Ej NN: j*D + e)
    gemm_f32_wmma<<<dim3(T/64, D/64, B*N), blk, 0, stream>>>(
        w2, LNTN, (long long)T*N, N,   Ej, (long long)N*D, 0, 1, D,   nullptr,
        out, LNTD, (long long)T*D, D, 1,   T, D, N, N, 1);

    // 14) z += per (b,t): sum_j w2[b,i,t,j]*Bm[b,j,t,e]   (Bm NN: j*(T*D) + e)
    gemm_f32_wmma<<<dim3(N/64, D/64, B*T), blk, 0, stream>>>(
        w2, LNTN, (long long)N, T*N,   Bm, LNTD, (long long)D, 1, T*D,   nullptr,
        out, LNTD, (long long)D, T*D, 1,   N, D, N, T, 1);

    // 15) wsum = sum_j w2
    wsum_kernel<<<B * N * T, blk, 0, stream>>>(w2, wsm);

    // 16) x = H + z + Aout*wsum ; LayerNorm  (in-place on d_out)
    ln_kernel<<<B * N * T, 128, 0, stream>>>(H, Aout, wsm, gam, bet, out);
}